// Extractor_52226802319863
// MI455X (gfx1250) — compile-verified
//
#include <hip/hip_runtime.h>

typedef float v2f __attribute__((ext_vector_type(2)));
typedef float v8f __attribute__((ext_vector_type(8)));

#define IMG_H 480
#define IMG_W 640
#define NPIX  (IMG_H * IMG_W)   // 307200
#define NBATCH 2
#define NRAY 9
#define VOX 256

// Output offsets (floats), concatenated flat in reference return order
#define OFF_EXT 0UL                               // (b,n,9)      5,529,600
#define OFF_RAY 5529600UL                         // (b,n,9,3)   16,588,800
#define OFF_DIR 22118400UL                        // (b,n,3)      1,843,200
#define OFF_DEP 23961600UL                        // (b,n)          614,400
#define OFF_IDX 24576000UL                        // (b,n,9,3)   16,588,800
#define OFF_CRD 41164800UL                        // (b,n,3)      1,843,200

// v_rcp_f32 + one Newton step (~1ulp); avoids the div_scale/div_fmas/div_fixup chain
__device__ __forceinline__ float fast_rcp(float x) {
    float r = __builtin_amdgcn_rcpf(x);
    r = r * (2.0f - x * r);
    return r;
}

__global__ __launch_bounds__(256)
void Extractor_52226802319863_kernel(const float* __restrict__ depth,
                                     const float* __restrict__ extr,
                                     const float* __restrict__ intr,
                                     const float* __restrict__ vol,
                                     const float* __restrict__ origin,
                                     const float* __restrict__ resolution,
                                     float* __restrict__ out)
{
    const int lane  = threadIdx.x & 31;
    const int wave  = threadIdx.x >> 5;
    const int chunk = blockIdx.x * (blockDim.x >> 5) + wave;   // 32-pixel chunk per wave
    const int g0    = chunk * 32;
    if (g0 >= NBATCH * NPIX) return;                           // wave-uniform guard

    const int g = g0 + lane;
    const int b = g0 / NPIX;                                   // chunks never straddle batches

    // ---- per-thread fused transform M[3][4] = [E3x3 * Kinv | t] (cheap, ~100 flops) ----
    const float* K = intr + 9 * b;
    const float* E = extr + 16 * b;
    float a00=K[0],a01=K[1],a02=K[2],a10=K[3],a11=K[4],a12=K[5],a20=K[6],a21=K[7],a22=K[8];
    float c00 =  a11*a22 - a12*a21;
    float c10 = -(a10*a22 - a12*a20);
    float c20 =  a10*a21 - a11*a20;
    float invdet = fast_rcp(a00*c00 + a01*c10 + a02*c20);
    float ki[9];
    ki[0] = c00*invdet;              ki[1] = (a02*a21-a01*a22)*invdet; ki[2] = (a01*a12-a02*a11)*invdet;
    ki[3] = c10*invdet;              ki[4] = (a00*a22-a02*a20)*invdet; ki[5] = (a02*a10-a00*a12)*invdet;
    ki[6] = c20*invdet;              ki[7] = (a01*a20-a00*a21)*invdet; ki[8] = (a00*a11-a01*a10)*invdet;
    float M[3][4];
    #pragma unroll
    for (int r = 0; r < 3; ++r) {
        #pragma unroll
        for (int c = 0; c < 3; ++c)
            M[r][c] = E[r*4+0]*ki[0*3+c] + E[r*4+1]*ki[1*3+c] + E[r*4+2]*ki[2*3+c];
        M[r][3] = E[r*4+3];
    }

    // ---- A operand: 16x4 f32 (rows 0..2 = M, rest zero) ----
    // lanes 0-15: M=lane, VGPR0=K0, VGPR1=K1 ; lanes 16-31: M=lane-16, VGPR0=K2, VGPR1=K3
    const int mrow  = lane & 15;
    const int kbase = (lane < 16) ? 0 : 2;
    v2f A;
    A[0] = (mrow < 3) ? M[mrow][kbase + 0] : 0.0f;
    A[1] = (mrow < 3) ? M[mrow][kbase + 1] : 0.0f;

    // ---- per-lane depth (read-once: non-temporal), B operands for the two halves ----
    const float z = __builtin_nontemporal_load(depth + g);
    v2f B[2];
    #pragma unroll
    for (int h = 0; h < 2; ++h) {
        int psel   = g0 + h * 16 + (lane & 15);        // pixel whose column this lane feeds
        int pixsel = psel - b * NPIX;
        float zsel = __shfl(z, h * 16 + (lane & 15), 32);
        float usel = (float)(pixsel % IMG_W);          // yy (col)
        float vsel = (float)(pixsel / IMG_W);          // xx (row)
        // B 4x16: VGPR0 = {K0 row | K2 row}, VGPR1 = {K1 row | K3 row}
        B[h][0] = (lane < 16) ? usel * zsel : zsel;    // K0 = u*z ; K2 = z
        B[h][1] = (lane < 16) ? vsel * zsel : 1.0f;    // K1 = v*z ; K3 = 1
    }

    // ---- two WMMAs transform all 32 pixels of the chunk ----
    v8f czero = {};
    v8f dLo = __builtin_amdgcn_wmma_f32_16x16x4_f32(false, A, false, B[0], (short)0, czero, false, false);
    v8f dHi = __builtin_amdgcn_wmma_f32_16x16x4_f32(false, A, false, B[1], (short)0, czero, false, false);

    // D rows 0..2 (world x,y,z) live in lanes 0..15; redistribute to all 32 lanes.
    // Shuffles executed unconditionally so source lanes are in EXEC.
    float x0 = __shfl(dLo[0], lane & 15, 32), x1 = __shfl(dHi[0], lane & 15, 32);
    float y0 = __shfl(dLo[1], lane & 15, 32), y1 = __shfl(dHi[1], lane & 15, 32);
    float z0 = __shfl(dLo[2], lane & 15, 32), z1 = __shfl(dHi[2], lane & 15, 32);
    const float wx = (lane < 16) ? x0 : x1;
    const float wy = (lane < 16) ? y0 : y1;
    const float wz = (lane < 16) ? z0 : z1;

    // ---- voxel-space ray setup ----
    const float org0 = origin[0], org1 = origin[1], org2 = origin[2];
    const float invres = fast_rcp(resolution[0]);
    const float cvx = (wx - org0) * invres;
    const float cvy = (wy - org1) * invres;
    const float cvz = (wz - org2) * invres;
    const float evx = (E[3]  - org0) * invres;
    const float evy = (E[7]  - org1) * invres;
    const float evz = (E[11] - org2) * invres;
    float dx = cvx - evx, dy = cvy - evy, dz = cvz - evz;
    const float nrm = sqrtf(dx*dx + dy*dy + dz*dz);
    const float rn  = fast_rcp(fmaxf(nrm, 1e-12f));
    dx *= rn; dy *= rn; dz *= rn;

    // ---- small per-pixel outputs (write-once: non-temporal, keep L2 for the volume) ----
    __builtin_nontemporal_store(z, out + OFF_DEP + (size_t)g);
    {
        size_t d3 = (size_t)g * 3;
        __builtin_nontemporal_store(dx, out + OFF_DIR + d3 + 0);
        __builtin_nontemporal_store(dy, out + OFF_DIR + d3 + 1);
        __builtin_nontemporal_store(dz, out + OFF_DIR + d3 + 2);
        __builtin_nontemporal_store(wx, out + OFF_CRD + d3 + 0);
        __builtin_nontemporal_store(wy, out + OFF_CRD + d3 + 1);
        __builtin_nontemporal_store(wz, out + OFF_CRD + d3 + 2);
    }

    // ---- 9-point ray walk: floor, clamp, L2-resident gather, NT stream out ----
    #pragma unroll
    for (int k = 0; k < NRAY; ++k) {
        const float off = (float)(k - 4);                     // BIN_SIZE = 1.0
        const float px = cvx + off * dx;
        const float py = cvy + off * dy;
        const float pz = cvz + off * dz;
        const int ix = (int)floorf(px);
        const int iy = (int)floorf(py);
        const int iz = (int)floorf(pz);
        const bool valid = (ix >= 0) & (ix < VOX) & (iy >= 0) & (iy < VOX) & (iz >= 0) & (iz < VOX);
        const int cx = min(max(ix, 0), VOX - 1);
        const int cy = min(max(iy, 0), VOX - 1);
        const int cz = min(max(iz, 0), VOX - 1);
        // regular (temporal) load: volume is 64MB and must stay L2-resident
        const float val = vol[((size_t)cx << 16) | ((size_t)cy << 8) | (size_t)cz];
        __builtin_nontemporal_store(valid ? val : 0.0f, out + OFF_EXT + (size_t)g * NRAY + k);
        const size_t rb = OFF_RAY + ((size_t)g * NRAY + k) * 3;
        __builtin_nontemporal_store(px, out + rb + 0);
        __builtin_nontemporal_store(py, out + rb + 1);
        __builtin_nontemporal_store(pz, out + rb + 2);
        const size_t ib = OFF_IDX + ((size_t)g * NRAY + k) * 3;
        __builtin_nontemporal_store((float)ix, out + ib + 0);
        __builtin_nontemporal_store((float)iy, out + ib + 1);
        __builtin_nontemporal_store((float)iz, out + ib + 2);
    }
}

extern "C" void kernel_launch(void* const* d_in, const int* in_sizes, int n_in,
                              void* d_out, int out_size, void* d_ws, size_t ws_size,
                              hipStream_t stream) {
    const float* depth = (const float*)d_in[0];
    const float* extr  = (const float*)d_in[1];
    const float* intr  = (const float*)d_in[2];
    const float* vol   = (const float*)d_in[3];
    const float* orig  = (const float*)d_in[4];
    const float* res   = (const float*)d_in[5];
    float* out = (float*)d_out;

    // 614,400 pixels / (32 per wave * 8 waves per block) = 2400 blocks, all waves full
    const int blocks = (NBATCH * NPIX) / (32 * 8);
    Extractor_52226802319863_kernel<<<blocks, 256, 0, stream>>>(depth, extr, intr, vol, orig, res, out);
}